// MLPPredictor_35682588295604
// MI455X (gfx1250) — compile-verified
//
#include <hip/hip_runtime.h>
#include <hip/hip_bf16.h>

// ---------------------------------------------------------------------------
// Edge scorer: out[e] = sigmoid(h[src[e]].Wu + h[dst[e]].Wv + b)
// Two-pass: (1) WMMA node projection pu/pv, (2) gather + sigmoid per edge.
// ---------------------------------------------------------------------------

typedef __attribute__((ext_vector_type(2))) float v2f;
typedef __attribute__((ext_vector_type(8))) float v8f;

#define D 128          // feature dim (in_sizes[3] == 2*D)
#define KSTEPS (D / 4) // 32 wmma_f32_16x16x4_f32 per 16-node tile

// ---------------------------------------------------------------------------
// Pass 1: per-node projections via V_WMMA_F32_16X16X4_F32.
// One wave handles one 16-node tile: C(16x16) = H(16x128) x W(128x16),
// where W col0 = Wu, col1 = Wv, cols 2..15 = 0.
// ---------------------------------------------------------------------------
__global__ __launch_bounds__(256) void node_proj_wmma(
    const float* __restrict__ h,   // [nNodes, 128]
    const float* __restrict__ Ww,  // [256] = {Wu[0:128], Wv[0:128]}
    float* __restrict__ pu,        // [nNodes]
    float* __restrict__ pv,        // [nNodes]
    int nNodes, int nTiles)
{
    const int lane  = threadIdx.x & 31;
    const int wavesPerBlock = blockDim.x >> 5;
    const int wave  = blockIdx.x * wavesPerBlock + (threadIdx.x >> 5);
    const int nWaves = gridDim.x * wavesPerBlock;

    const int col   = lane & 15;        // N index for B/C fragments, M row for A
    const int khalf = (lane >> 4) * 2;  // K sub-offset: lanes 0-15 -> {0,1}, 16-31 -> {2,3}

    // --- Preload B fragments for all 32 K-steps (kept in VGPRs, reused). ---
    // B(4x16) layout assumed to mirror A: VGPR0 = K=khalf, VGPR1 = K=khalf+1,
    // N = lane&15.  Only N==0 (Wu) and N==1 (Wv) are non-zero.
    v2f bfrag[KSTEPS];
    #pragma unroll
    for (int kk = 0; kk < KSTEPS; ++kk) {
        float b0 = 0.0f, b1 = 0.0f;
        if (col < 2) {
            const float* wc = Ww + col * D + kk * 4 + khalf;
            b0 = wc[0];
            b1 = wc[1];
        }
        bfrag[kk].x = b0;
        bfrag[kk].y = b1;
    }

    // --- Grid-stride over 16-node tiles (wave-uniform, EXEC stays all-1). ---
    for (int tile = wave; tile < nTiles; tile += nWaves) {
        int row = tile * 16 + col;
        if (row >= nNodes) row = nNodes - 1;   // clamped duplicate load (tail)
        const float* hp = h + (size_t)row * D + khalf;

        v8f c = {};
        #pragma unroll
        for (int kk = 0; kk < KSTEPS; ++kk) {
            // A(16x4): lane holds h[row][kk*4 + khalf + {0,1}] -> global_load_b64
            v2f a = *(const v2f*)(hp + kk * 4);
            // 8 args: (neg_a, A, neg_b, B, c_mod, C, reuse_a, reuse_b)
            c = __builtin_amdgcn_wmma_f32_16x16x4_f32(
                    false, a, false, bfrag[kk], (short)0, c, false, false);
        }

        // --- Extract C columns 0 (pu) and 1 (pv). ---
        // C layout: VGPR r -> lanes 0-15: (M=r, N=lane); lanes 16-31: (M=r+8, N=lane-16)
        if (col < 2) {
            float* op = (col == 0) ? pu : pv;
            const int base = tile * 16 + (lane >> 4) * 8;
            #pragma unroll
            for (int r = 0; r < 8; ++r) {
                const int idx = base + r;
                if (idx < nNodes) op[idx] = c[r];
            }
        }
    }
}

// ---------------------------------------------------------------------------
// Pass 2: edge gather + sigmoid.  Indices coalesced via int4; pu/pv tables
// (400 KB each) are L2-resident on MI455X (192 MB L2).
// ---------------------------------------------------------------------------
__device__ __forceinline__ float sigmoidf(float x) {
    return 1.0f / (1.0f + __expf(-x));
}

__global__ __launch_bounds__(256) void edge_score(
    const int* __restrict__ src,
    const int* __restrict__ dst,
    const float* __restrict__ pu,
    const float* __restrict__ pv,
    const float* __restrict__ bias,
    float* __restrict__ out,
    int nE)
{
    const int i  = blockIdx.x * blockDim.x + threadIdx.x;
    const int nv = nE >> 2;           // number of int4 groups
    const float b = bias[0];

    if (i < nv) {
        const int4 s4 = ((const int4*)src)[i];
        const int4 d4 = ((const int4*)dst)[i];
        float4 r;
        r.x = sigmoidf(pu[s4.x] + pv[d4.x] + b);
        r.y = sigmoidf(pu[s4.y] + pv[d4.y] + b);
        r.z = sigmoidf(pu[s4.z] + pv[d4.z] + b);
        r.w = sigmoidf(pu[s4.w] + pv[d4.w] + b);
        ((float4*)out)[i] = r;
    }
    // scalar tail (nE not a multiple of 4)
    const int t = nv * 4 + i;
    if (i < (nE & 3)) {
        out[t] = sigmoidf(pu[src[t]] + pv[dst[t]] + b);
    }
}

// ---------------------------------------------------------------------------
// Launcher
// ---------------------------------------------------------------------------
extern "C" void kernel_launch(void* const* d_in, const int* in_sizes, int n_in,
                              void* d_out, int out_size, void* d_ws, size_t ws_size,
                              hipStream_t stream)
{
    const float* h    = (const float*)d_in[0];  // [nNodes*128]
    const int*   src  = (const int*)  d_in[1];  // [4*400000] flattened
    const int*   dst  = (const int*)  d_in[2];
    const float* Ww   = (const float*)d_in[3];  // [256]
    const float* Wb   = (const float*)d_in[4];  // [1]
    float*       out  = (float*)d_out;

    const int nNodes = in_sizes[0] / D;
    const int nE     = in_sizes[1];             // flat edge count (src)
    const int nTiles = (nNodes + 15) / 16;

    float* pu = (float*)d_ws;
    float* pv = pu + nNodes;

    // Pass 1: one wave per 16-node tile, 8 waves per 256-thread block.
    const int wavesPerBlock = 256 / 32;
    int blocks1 = (nTiles + wavesPerBlock - 1) / wavesPerBlock;
    if (blocks1 > 4096) blocks1 = 4096;         // grid-stride covers the rest
    if (blocks1 < 1) blocks1 = 1;
    node_proj_wmma<<<blocks1, 256, 0, stream>>>(h, Ww, pu, pv, nNodes, nTiles);

    // Pass 2: vectorized edge pass (same stream -> ordered after pass 1).
    int nv = nE >> 2;
    int blocks2 = (nv + 255) / 256;
    if (blocks2 < 1) blocks2 = 1;
    edge_score<<<blocks2, 256, 0, stream>>>(src, dst, pu, pv, Wb, out, nE);
}